// ND_Unitary_1194000908844
// MI455X (gfx1250) — compile-verified
//
#include <hip/hip_runtime.h>

// CDNA5 / gfx1250, wave32. ND_Unitary via chunked Givens-rotation transforms
// applied with V_WMMA_F32_16X16X4_F32.

typedef float v2f __attribute__((ext_vector_type(2)));
typedef float v8f __attribute__((ext_vector_type(8)));

constexpr int N_DIM    = 1024;
constexpr int K_ANG    = N_DIM * (N_DIM - 1) / 2;   // 523,776 angles
constexpr int T_STRIDE = 17 * 17;                   // one chunk transform

// #chunks for stages 2..n-1 (prefix sum of ceil((m-1)/16)).
__device__ __host__ __forceinline__ int chunk_base_for_stage(int n) {
    int v = n - 2;                 // f(v) = sum_{u=1..v} ceil(u/16)
    int q = v >> 4, r = v & 15;
    return 8 * q * (q + 1) + r * (q + 1);
}

// ---------------------------------------------------------------- kernel 1 --
// cos/sin of every angle (massively parallel).
__global__ void cs_kernel(const float* __restrict__ alpha,
                          float* __restrict__ cws, float* __restrict__ sws) {
    int k = blockIdx.x * blockDim.x + threadIdx.x;
    if (k < K_ANG) {
        float a = alpha[k];
        cws[k] = cosf(a);
        sws[k] = sinf(a);
    }
}

// ---------------------------------------------------------------- kernel 2 --
// Build the 17x17 transform of each 16-rotation chunk. Basis = (16 block
// rows i0..i0+15, last). Row k of T is finalized at step k (rows > k are
// untouched until their own rotation), so we only carry the evolving last
// row L. Padded slots (i > n-2) are identity rotations.
__global__ __launch_bounds__(32)
void build_T_kernel(const float* __restrict__ cws, const float* __restrict__ sws,
                    float* __restrict__ Tws) {
    const int n   = (int)blockIdx.y + 2;       // stage 2..N
    const int cis = (int)blockIdx.x;           // chunk-in-stage
    const int i0  = cis << 4;
    if (i0 > n - 2) return;
    const int col = threadIdx.x;               // basis column 0..16
    if (col >= 17) return;

    float* __restrict__ T = Tws + (size_t)(chunk_base_for_stage(n) + cis) * T_STRIDE;
    const int abase = ((n - 1) * (n - 2)) >> 1;

    float L = (col == 16) ? 1.0f : 0.0f;       // last-row accumulator
    #pragma unroll
    for (int k = 0; k < 16; ++k) {
        const int i = i0 + k;
        float cc = 1.0f, ss = 0.0f;
        if (i <= n - 2) { cc = cws[abase + i]; ss = sws[abase + i]; }
        const float dlt = (col == k) ? 1.0f : 0.0f;
        T[k * 17 + col] = cc * dlt - ss * L;   // finalized row k
        L = ss * dlt + cc * L;
    }
    T[16 * 17 + col] = L;                      // new last row
}

// ---------------------------------------------------------------- kernel 3 --
// One wave owns 16 columns of M, resident in LDS (1024x16 f32 = 64 KB).
// For every (stage, chunk): block' = T[:16,:16]@block + T[:16,16] (x) last,
// last' = T[16,:16]@block + T[16,16]*last.  The 16x16 matmul is 4 chained
// V_WMMA_F32_16X16X4_F32 ops. No inter-wave communication at all.
__global__ __launch_bounds__(32)
void apply_kernel(const float* __restrict__ Tws, float* __restrict__ out) {
    __shared__ float Mc[N_DIM * 16];           // column slice, row-major x16
    const int lane    = threadIdx.x;           // 0..31 (wave32)
    const int half    = lane >> 4;             // lane half selects K/M offset
    const int c       = lane & 15;             // column within slice / N index
    const int koff    = half * 2;
    const int colBase = (int)blockIdx.x * 16;

    for (int idx = lane; idx < N_DIM * 16; idx += 32) {
        int r = idx >> 4, cc = idx & 15;
        Mc[idx] = (r == colBase + cc) ? 1.0f : 0.0f;   // M = I
    }
    __syncthreads();

    int chunkBase = 0;
    for (int n = 2; n <= N_DIM; ++n) {
        const int nch     = (n + 14) >> 4;     // ceil((n-1)/16)
        const int lastRow = n - 1;
        for (int ch = 0; ch < nch; ++ch) {
            const float* __restrict__ T = Tws + (size_t)(chunkBase + ch) * T_STRIDE;
            const int i0 = ch << 4;
            __builtin_prefetch(T + T_STRIDE, 0, 1);    // next chunk's transform

            // A operand: T[m][kbase+koff+{0,1}], m = lane&15 (ISA 16x4 layout)
            const float* Trow = T + c * 17 + koff;
            v2f a0, a1, a2, a3;
            a0.x = Trow[0];  a0.y = Trow[1];
            a1.x = Trow[4];  a1.y = Trow[5];
            a2.x = Trow[8];  a2.y = Trow[9];
            a3.x = Trow[12]; a3.y = Trow[13];

            // B operand from LDS: block[kbase+koff+{0,1}][c] (4x16 layout)
            const int rb = (i0 + koff) * 16 + c;
            v2f b0, b1, b2, b3;
            b0.x = Mc[rb +  0 * 16]; b0.y = Mc[rb +  1 * 16];
            b1.x = Mc[rb +  4 * 16]; b1.y = Mc[rb +  5 * 16];
            b2.x = Mc[rb +  8 * 16]; b2.y = Mc[rb +  9 * 16];
            b3.x = Mc[rb + 12 * 16]; b3.y = Mc[rb + 13 * 16];

            const float lastval = Mc[lastRow * 16 + c];

            // last' = T[16,:16]@block + T[16][16]*last, using B regs + xor-16
            const float* T16 = T + 16 * 17;
            float part =
                b0.x * T16[koff +  0] + b0.y * T16[koff +  1] +
                b1.x * T16[koff +  4] + b1.y * T16[koff +  5] +
                b2.x * T16[koff +  8] + b2.y * T16[koff +  9] +
                b3.x * T16[koff + 12] + b3.y * T16[koff + 13];
            part += __shfl_xor(part, 16, 32);
            const float nl = part + T16[16] * lastval;

            // D = T[:16,:16] @ block  (4 accumulating K=4 WMMAs)
            v8f d = {0.f, 0.f, 0.f, 0.f, 0.f, 0.f, 0.f, 0.f};
            d = __builtin_amdgcn_wmma_f32_16x16x4_f32(false, a0, false, b0, (short)0, d, false, false);
            d = __builtin_amdgcn_wmma_f32_16x16x4_f32(false, a1, false, b1, (short)0, d, false, false);
            d = __builtin_amdgcn_wmma_f32_16x16x4_f32(false, a2, false, b2, (short)0, d, false, false);
            d = __builtin_amdgcn_wmma_f32_16x16x4_f32(false, a3, false, b3, (short)0, d, false, false);

            // rank-1 correction: d[v] += T[row][16] * last
            #pragma unroll
            for (int v = 0; v < 8; ++v)
                d[v] += T[(v + 8 * half) * 17 + 16] * lastval;

            __builtin_amdgcn_wave_barrier();   // reads above, writes below

            #pragma unroll
            for (int v = 0; v < 8; ++v)
                Mc[(i0 + v + 8 * half) * 16 + c] = d[v];   // block rows
            if (half == 0)
                Mc[lastRow * 16 + c] = nl;                 // then last row
            __builtin_amdgcn_wave_barrier();
        }
        chunkBase += nch;
    }
    __syncthreads();

    for (int idx = lane; idx < N_DIM * 16; idx += 32) {
        int r = idx >> 4, cc = idx & 15;
        out[(size_t)r * N_DIM + colBase + cc] = Mc[idx];
    }
}

// -------------------------------------------------------------------- host --
extern "C" void kernel_launch(void* const* d_in, const int* in_sizes, int n_in,
                              void* d_out, int out_size, void* d_ws, size_t ws_size,
                              hipStream_t stream) {
    (void)in_sizes; (void)n_in; (void)out_size; (void)ws_size;
    const float* alpha = (const float*)d_in[0];   // d_in[1] = N (fixed 1024)

    // Workspace layout (floats): [cos K][sin K][T: 33216 * 289] ~= 42.6 MB
    float* ws  = (float*)d_ws;
    float* cws = ws;
    float* sws = ws + K_ANG;
    float* Tws = ws + 2 * (size_t)K_ANG;

    cs_kernel<<<(K_ANG + 255) / 256, 256, 0, stream>>>(alpha, cws, sws);

    dim3 g2(64, N_DIM - 1, 1);                    // (max chunks, stages)
    build_T_kernel<<<g2, 32, 0, stream>>>(cws, sws, Tws);

    apply_kernel<<<N_DIM / 16, 32, 0, stream>>>(Tws, (float*)d_out);
}